// PLGCN_57844619543132
// MI455X (gfx1250) — compile-verified
//
#include <hip/hip_runtime.h>

#define N_NODES 100000
#define N_EDGES 1600000
#define F_IN    256
#define F_HID   64
#define F_OUT   40

typedef __attribute__((ext_vector_type(2))) float v2f;
typedef __attribute__((ext_vector_type(8))) float v8f;

// ---------------------------------------------------------------------------
// Degree / normalization: deg[i] = 1 (self loop) + sum_{e: col==i} ew[e]
// dis[i] = rsqrt(deg[i])
// ---------------------------------------------------------------------------
__global__ void k_deg_init(float* __restrict__ deg) {
    int i = blockIdx.x * blockDim.x + threadIdx.x;
    if (i < N_NODES) deg[i] = 1.0f;   // self-loop weight
}

__global__ void k_deg_scatter(const long long* __restrict__ ei,
                              const float* __restrict__ ew,
                              float* __restrict__ deg) {
    int e = blockIdx.x * blockDim.x + threadIdx.x;
    if (e < N_EDGES) {
        int col = (int)ei[(size_t)N_EDGES + e];  // edge_index[1][e]
        atomicAdd(&deg[col], ew[e]);
    }
}

__global__ void k_rsqrt(float* __restrict__ dis) {
    int i = blockIdx.x * blockDim.x + threadIdx.x;
    if (i < N_NODES) {
        float d = dis[i];
        dis[i] = (d > 0.0f) ? rsqrtf(fmaxf(d, 1e-12f)) : 0.0f;
    }
}

// ---------------------------------------------------------------------------
// GEMM1: h1 = x @ W1   [100000 x 256] @ [256 x 64]
// One wave per 16x16 output tile; K stepped by 4 with V_WMMA_F32_16X16X4_F32.
// fp32 A layout (16x4): lanes 0-15 hold M=0..15 K={k,k+1}; lanes 16-31 K={k+2,k+3}
// fp32 B/C layout: row striped across lanes within a VGPR (halves at M+8 / K+2)
// ---------------------------------------------------------------------------
__global__ __launch_bounds__(256) void k_gemm1(const float* __restrict__ X,
                                               const float* __restrict__ W,
                                               float* __restrict__ H) {
    const int wave = (int)((blockIdx.x * blockDim.x + threadIdx.x) >> 5);
    const int lane = threadIdx.x & 31;
    const int rt = wave >> 2;            // 6250 row tiles
    const int ct = wave & 3;             // 4 col tiles
    if (rt >= N_NODES / 16) return;      // uniform per wave: EXEC stays all-ones
    const int row0 = rt * 16, col0 = ct * 16;
    const int half = lane >> 4;          // 0: lanes 0-15, 1: lanes 16-31
    const int l    = lane & 15;

    const float* __restrict__ xrow = X + (size_t)(row0 + l) * F_IN;
    v8f c = {};
    #pragma unroll 4
    for (int k = 0; k < F_IN; k += 4) {
        const int ka = k + half * 2;
        v2f a, b;
        a.x = xrow[ka];
        a.y = xrow[ka + 1];
        b.x = W[(size_t)ka * F_HID + col0 + l];
        b.y = W[(size_t)(ka + 1) * F_HID + col0 + l];
        c = __builtin_amdgcn_wmma_f32_16x16x4_f32(false, a, false, b,
                                                  (short)0, c, false, false);
    }
    #pragma unroll
    for (int v = 0; v < 8; ++v) {
        const int m = v + half * 8;
        H[(size_t)(row0 + m) * F_HID + col0 + l] = c[v];
    }
}

// ---------------------------------------------------------------------------
// Aggregation layer 1: out1 = b1 + dis[i]^2 * h1[i] + sum_e norm_e * h1[row_e]
// ---------------------------------------------------------------------------
__global__ void k_agg1_init(const float* __restrict__ H,
                            const float* __restrict__ dis,
                            const float* __restrict__ b1,
                            float* __restrict__ OUT) {
    size_t idx = (size_t)blockIdx.x * blockDim.x + threadIdx.x;
    if (idx < (size_t)N_NODES * F_HID) {
        int i = (int)(idx >> 6);       // /64
        int f = (int)(idx & 63);
        float d = dis[i];
        OUT[idx] = b1[f] + d * d * H[idx];
    }
}

__global__ void k_agg1_edges(const long long* __restrict__ ei,
                             const float* __restrict__ ew,
                             const float* __restrict__ dis,
                             const float* __restrict__ H,
                             float* __restrict__ OUT) {
    size_t idx = (size_t)blockIdx.x * blockDim.x + threadIdx.x;
    size_t e = idx >> 6;               // 64 threads per edge
    int   f  = (int)(idx & 63);
    if (e < N_EDGES) {
        int row = (int)ei[e];
        int col = (int)ei[(size_t)N_EDGES + e];
        float norm = dis[row] * ew[e] * dis[col];
        atomicAdd(&OUT[(size_t)col * F_HID + f],
                  norm * H[(size_t)row * F_HID + f]);
    }
}

// ---------------------------------------------------------------------------
// GEMM2: h2 = relu(out1) @ W2   [100000 x 64] @ [64 x 40] (3 col tiles, guard 40)
// ---------------------------------------------------------------------------
__global__ __launch_bounds__(256) void k_gemm2(const float* __restrict__ A,
                                               const float* __restrict__ W,
                                               float* __restrict__ H) {
    const int wave = (int)((blockIdx.x * blockDim.x + threadIdx.x) >> 5);
    const int lane = threadIdx.x & 31;
    const int rt = wave / 3;
    const int ct = wave % 3;             // 3 col tiles covering 40 cols
    if (rt >= N_NODES / 16) return;      // uniform per wave
    const int row0 = rt * 16, col0 = ct * 16;
    const int half = lane >> 4;
    const int l    = lane & 15;
    const int n    = col0 + l;           // output column this lane owns (B/C layout)
    const bool nok = (n < F_OUT);

    const float* __restrict__ arow = A + (size_t)(row0 + l) * F_HID;
    v8f c = {};
    #pragma unroll
    for (int k = 0; k < F_HID; k += 4) {
        const int ka = k + half * 2;
        v2f a, b;
        a.x = fmaxf(arow[ka], 0.0f);       // fused ReLU
        a.y = fmaxf(arow[ka + 1], 0.0f);
        b.x = nok ? W[(size_t)ka * F_OUT + n] : 0.0f;
        b.y = nok ? W[(size_t)(ka + 1) * F_OUT + n] : 0.0f;
        c = __builtin_amdgcn_wmma_f32_16x16x4_f32(false, a, false, b,
                                                  (short)0, c, false, false);
    }
    if (nok) {
        #pragma unroll
        for (int v = 0; v < 8; ++v) {
            const int m = v + half * 8;
            H[(size_t)(row0 + m) * F_OUT + n] = c[v];
        }
    }
}

// ---------------------------------------------------------------------------
// Aggregation layer 2 -> d_out
// ---------------------------------------------------------------------------
__global__ void k_agg2_init(const float* __restrict__ H,
                            const float* __restrict__ dis,
                            const float* __restrict__ b2,
                            float* __restrict__ OUT) {
    size_t idx = (size_t)blockIdx.x * blockDim.x + threadIdx.x;
    if (idx < (size_t)N_NODES * F_OUT) {
        unsigned i = (unsigned)(idx / F_OUT);
        unsigned f = (unsigned)(idx % F_OUT);
        float d = dis[i];
        OUT[idx] = b2[f] + d * d * H[idx];
    }
}

__global__ void k_agg2_edges(const long long* __restrict__ ei,
                             const float* __restrict__ ew,
                             const float* __restrict__ dis,
                             const float* __restrict__ H,
                             float* __restrict__ OUT) {
    size_t idx = (size_t)blockIdx.x * blockDim.x + threadIdx.x;
    if (idx < (size_t)N_EDGES * F_OUT) {
        unsigned e = (unsigned)(idx / F_OUT);
        unsigned f = (unsigned)(idx % F_OUT);
        int row = (int)ei[e];
        int col = (int)ei[(size_t)N_EDGES + e];
        float norm = dis[row] * ew[e] * dis[col];
        atomicAdd(&OUT[(size_t)col * F_OUT + f],
                  norm * H[(size_t)row * F_OUT + f]);
    }
}

// ---------------------------------------------------------------------------
extern "C" void kernel_launch(void* const* d_in, const int* in_sizes, int n_in,
                              void* d_out, int out_size, void* d_ws, size_t ws_size,
                              hipStream_t stream) {
    const float*      x  = (const float*)d_in[0];
    const long long*  ei = (const long long*)d_in[1];
    const float*      ew = (const float*)d_in[2];
    const float*      W1 = (const float*)d_in[3];
    const float*      b1 = (const float*)d_in[4];
    const float*      W2 = (const float*)d_in[5];
    const float*      b2 = (const float*)d_in[6];
    float*            out = (float*)d_out;

    // Workspace layout (bytes), 256-aligned:
    //   dis : N floats
    //   h1  : N*64 floats   (reused as h2 (N*40) after h1 goes dead)
    //   out1: N*64 floats
    char* ws = (char*)d_ws;
    size_t o_dis  = 0;
    size_t o_h1   = ((size_t)N_NODES * 4 + 255) & ~(size_t)255;
    size_t o_out1 = o_h1 + ((size_t)N_NODES * F_HID * 4 + 255) & ~(size_t)255;
    float* dis  = (float*)(ws + o_dis);
    float* h1   = (float*)(ws + o_h1);
    float* out1 = (float*)(ws + o_out1);
    float* h2   = h1;   // reuse: h1 dead once gemm2 consumes relu(out1)

    const int B = 256;

    // 1) degree + rsqrt
    k_deg_init   <<<(N_NODES + B - 1) / B, B, 0, stream>>>(dis);
    k_deg_scatter<<<(N_EDGES + B - 1) / B, B, 0, stream>>>(ei, ew, dis);
    k_rsqrt      <<<(N_NODES + B - 1) / B, B, 0, stream>>>(dis);

    // 2) h1 = x @ W1 : 6250*4 = 25000 tiles, 8 waves/block
    k_gemm1<<<(25000 + 7) / 8, B, 0, stream>>>(x, W1, h1);

    // 3) layer-1 aggregation
    k_agg1_init <<<(int)(((size_t)N_NODES * F_HID + B - 1) / B), B, 0, stream>>>(h1, dis, b1, out1);
    k_agg1_edges<<<(int)(((size_t)N_EDGES * F_HID + B - 1) / B), B, 0, stream>>>(ei, ew, dis, h1, out1);

    // 4) h2 = relu(out1) @ W2 : 6250*3 = 18750 tiles
    k_gemm2<<<(18750 + 7) / 8, B, 0, stream>>>(out1, W2, h2);

    // 5) layer-2 aggregation -> d_out
    k_agg2_init <<<(int)(((size_t)N_NODES * F_OUT + B - 1) / B), B, 0, stream>>>(h2, dis, b2, out);
    k_agg2_edges<<<(int)(((size_t)N_EDGES * F_OUT + B - 1) / B), B, 0, stream>>>(ei, ew, dis, h2, out);

    (void)in_sizes; (void)n_in; (void)out_size; (void)ws_size;
}